// Attention_49349174231422
// MI455X (gfx1250) — compile-verified
//
#include <hip/hip_runtime.h>
#include <hip/hip_bf16.h>

// Problem constants (from reference)
#define B_   64
#define S_   256
#define DIM_ 512
#define H_   16
#define DH_  32            // DIM/H
#define MTOT (B_ * S_)     // 16384 rows in QKV GEMM
#define QKV_ELEMS ((size_t)B_ * H_ * S_ * DH_)  // 8,388,608 floats per tensor

typedef __attribute__((ext_vector_type(2))) float v2f;
typedef __attribute__((ext_vector_type(8))) float v8f;

typedef __attribute__((address_space(1))) int as1_int;
typedef __attribute__((address_space(3))) int as3_int;

__device__ __forceinline__ v8f wmma_f32_k4(v2f a, v2f b, v8f c) {
    // V_WMMA_F32_16X16X4_F32 : D = A(16x4, f32) x B(4x16, f32) + C(16x16, f32)
    return __builtin_amdgcn_wmma_f32_16x16x4_f32(false, a, false, b, (short)0, c,
                                                 false, false);
}

// ---------------------------------------------------------------------------
// Kernel 1: fused QKV projection + bias + RoPE, output layout (B, H, S, DH).
// One wave32 computes one 16x16 output tile; K=512 -> 128 WMMA k4 steps.
// The 512x16 weight tile (shared by all 8 waves of the block) is staged into
// LDS once via async global->LDS copies (ASYNCcnt / s_wait_asynccnt), stored
// transposed+padded so the B fragment is one conflict-free ds_load_b64.
// grid = (128, 32, 3): 8 M-tiles per block (one per wave), 32 N-tiles, 3 mats.
// ---------------------------------------------------------------------------
__global__ void __launch_bounds__(256)
qkv_rope_kernel(const float* __restrict__ X,
                const float* __restrict__ Wq, const float* __restrict__ bq,
                const float* __restrict__ Wk, const float* __restrict__ bk,
                const float* __restrict__ Wv, const float* __restrict__ bv,
                float* __restrict__ outq, float* __restrict__ outk,
                float* __restrict__ outv) {
    __shared__ float wt[16][516];   // transposed weight tile, padded (33 KB)

    const int lane = threadIdx.x & 31;
    const int wave = threadIdx.x >> 5;
    const int row  = lane & 15;       // M (for A) / N (for B) index within tile
    const int half = lane >> 4;       // selects K=0,1 vs K=2,3 of the k4 step

    const int mtile = blockIdx.x * 8 + wave;   // 0..1023
    const int m0 = mtile * 16;
    const int n0 = blockIdx.y * 16;            // 0..511 step 16
    const int mat = blockIdx.z;                // 0=Q 1=K 2=V

    const float* Wm   = (mat == 0) ? Wq : (mat == 1) ? Wk : Wv;
    const float* bias = (mat == 0) ? bq : (mat == 1) ? bk : bv;
    float*       dst  = (mat == 0) ? outq : (mat == 1) ? outk : outv;

    // ---- Stage W[0:512, n0:n0+16] into LDS transposed: wt[c][k] = W[k][n0+c].
    // 256 lanes x 32 iters x 4B async copies; lanes 0..15 read 64B contiguous.
    {
        const int tid = threadIdx.x;
        const int c  = tid & 15;
        const int kb = tid >> 4;          // 0..15
#pragma unroll
        for (int it = 0; it < 32; ++it) {
            const int k = it * 16 + kb;
            const float* g = Wm + (size_t)k * DIM_ + n0 + c;
            float* l = &wt[c][k];
            __builtin_amdgcn_global_load_async_to_lds_b32(
                (as1_int*)(void*)g, (as3_int*)(void*)l,
                0 /*offset*/, 0 /*cpol*/);
        }
        asm volatile("s_wait_asynccnt 0x0" ::: "memory");
        __syncthreads();
    }

    // Per-lane A base pointer (float2: K=0,1 for lanes 0-15; K=2,3 for 16-31).
    const float* aptr = X + (size_t)(m0 + row) * DIM_ + 2 * half;

    v8f acc = {};
#pragma unroll 8
    for (int kk = 0; kk < DIM_; kk += 4) {
        v2f a = *(const v2f*)(aptr + kk);
        v2f b = *(const v2f*)&wt[row][kk + 2 * half];   // ds_load_b64
        acc = wmma_f32_k4(a, b, acc);
    }

    // Epilogue: bias, RoPE (Q,K only), scatter to (B,H,S,DH).
    const int n = n0 + row;        // global output column
    const int h = n >> 5;          // head
    const int d = n & 31;          // dim-in-head
    const int i = d >> 1;          // rotation pair index
    const float bval = bias[n];
    // inv_freq = 10000^(-i/16) = exp(-i * ln(10000)/16)
    const float inv_freq = __expf(-(float)i * (9.210340371976184f / 16.0f));

#pragma unroll
    for (int j = 0; j < 8; ++j) {
        const int m = m0 + j + half * 8;   // global row in [0, B*S)
        const int s = m & (S_ - 1);
        const int b_idx = m >> 8;
        float val = acc[j] + bval;
        if (mat < 2) {
            const float ang = (float)s * inv_freq;
            const float sv = __sinf(ang);
            const float cv = __cosf(ang);
            // partner column n^1 lives in lane^1 (same m: m depends on j,half only)
            const float other = __shfl_xor(val, 1, 32);
            val = (d & 1) ? (other * sv + val * cv)    // odd lane: x1*sin + x2*cos
                          : (val * cv - other * sv);   // even lane: x1*cos - x2*sin
        }
        dst[(((size_t)b_idx * H_ + h) * S_ + s) * DH_ + d] = val;
    }
}

// ---------------------------------------------------------------------------
// Kernel 2: attention for one (b, h, 16-query-row tile) per block.
//   scores = Q Kt / sqrt(DH)  (WMMA, staged to LDS)
//   probs  = softmax(scores) + rel_bias   (bias added AFTER softmax, per ref)
//   ctx    = probs @ V        (WMMA, A-fragment straight from LDS)
// grid = B*H*16 = 16384 blocks of 256 threads (8 waves).
// ---------------------------------------------------------------------------
__global__ void __launch_bounds__(256)
attn_kernel(const float* __restrict__ Qt, const float* __restrict__ Kt,
            const float* __restrict__ Vt, const float* __restrict__ bias_table,
            float* __restrict__ out) {
    __shared__ float sc[16 * S_];        // 16 q rows x 256 k cols = 16 KB
    __shared__ float part[8 * 16 * 16];  // per-wave ctx partials = 8 KB

    const int lane = threadIdx.x & 31;
    const int wave = threadIdx.x >> 5;
    const int row  = lane & 15;
    const int half = lane >> 4;

    const int bh = blockIdx.x >> 4;          // b*H + h
    const int qt = blockIdx.x & 15;          // q tile index
    const int h  = bh & (H_ - 1);
    const int b  = bh >> 4;
    const int s0 = qt * 16;

    const float* Q = Qt + (size_t)bh * S_ * DH_;
    const float* K = Kt + (size_t)bh * S_ * DH_;
    const float* V = Vt + (size_t)bh * S_ * DH_;

    // ---- Phase A: score tiles. Wave w handles k-tiles {w, w+8}. ----
    const float scale = 0.17677669529663687f;  // 1/sqrt(32)
#pragma unroll
    for (int t = 0; t < 2; ++t) {
        const int kt = wave + t * 8;
        const int n0 = kt * 16;
        v8f c = {};
#pragma unroll
        for (int kk = 0; kk < DH_; kk += 4) {
            v2f a = *(const v2f*)(Q + (size_t)(s0 + row) * DH_ + kk + 2 * half);
            v2f bb = *(const v2f*)(K + (size_t)(n0 + row) * DH_ + kk + 2 * half);
            c = wmma_f32_k4(a, bb, c);
        }
#pragma unroll
        for (int j = 0; j < 8; ++j)
            sc[(j + 8 * half) * S_ + n0 + row] = c[j] * scale;
    }
    __syncthreads();

    // ---- Phase B: softmax per row, then add relative bias. Wave w: rows 2w,2w+1. ----
#pragma unroll
    for (int rr = 0; rr < 2; ++rr) {
        const int r = wave * 2 + rr;
        float* rowp = sc + r * S_;
        float x[8];
        float mx = -3.0e38f;
#pragma unroll
        for (int cidx = 0; cidx < 8; ++cidx) {
            x[cidx] = rowp[lane + 32 * cidx];
            mx = fmaxf(mx, x[cidx]);
        }
#pragma unroll
        for (int off = 16; off >= 1; off >>= 1)
            mx = fmaxf(mx, __shfl_xor(mx, off, 32));
        float sum = 0.0f;
#pragma unroll
        for (int cidx = 0; cidx < 8; ++cidx) {
            x[cidx] = __expf(x[cidx] - mx);
            sum += x[cidx];
        }
#pragma unroll
        for (int off = 16; off >= 1; off >>= 1)
            sum += __shfl_xor(sum, off, 32);
        const float inv = 1.0f / sum;
        const int qg = s0 + r;
#pragma unroll
        for (int cidx = 0; cidx < 8; ++cidx) {
            const int col = lane + 32 * cidx;
            const int rel = qg - col + (S_ - 1);           // in [0, 510]
            rowp[col] = x[cidx] * inv + bias_table[rel * H_ + h];
        }
    }
    __syncthreads();

    // ---- Phase C: ctx = probs(16x256) @ V(256x32). ----
    // Wave w: n-tile = w&1 (d offset 0/16), K-slice = (w>>1)*64.
    {
        const int nt = wave & 1;
        const int ks = wave >> 1;
        const int n0 = nt * 16;
        v8f c = {};
#pragma unroll
        for (int kk = ks * 64; kk < ks * 64 + 64; kk += 4) {
            v2f a = *(const v2f*)(sc + row * S_ + kk + 2 * half);  // LDS ds_load_b64
            v2f bb;
            bb.x = V[(size_t)(kk + 2 * half) * DH_ + n0 + row];
            bb.y = V[(size_t)(kk + 2 * half + 1) * DH_ + n0 + row];
            c = wmma_f32_k4(a, bb, c);
        }
#pragma unroll
        for (int j = 0; j < 8; ++j)
            part[wave * 256 + (j + 8 * half) * 16 + row] = c[j];
    }
    __syncthreads();

    // Reduce 4 K-slice partials and write out[b, s, h*32 + d].
    const int tid = threadIdx.x;
#pragma unroll
    for (int e = 0; e < 2; ++e) {
        const int idx = tid * 2 + e;         // 0..511 over 16x32
        const int r  = idx >> 5;
        const int dd = idx & 31;
        const int nt = dd >> 4;
        const int col = dd & 15;
        float sum = 0.0f;
#pragma unroll
        for (int ks = 0; ks < 4; ++ks)
            sum += part[(ks * 2 + nt) * 256 + r * 16 + col];
        out[((size_t)b * S_ + s0 + r) * DIM_ + h * DH_ + dd] = sum;
    }
}

// ---------------------------------------------------------------------------
extern "C" void kernel_launch(void* const* d_in, const int* in_sizes, int n_in,
                              void* d_out, int out_size, void* d_ws, size_t ws_size,
                              hipStream_t stream) {
    const float* hidden     = (const float*)d_in[0];
    const float* Wq         = (const float*)d_in[1];
    const float* bq         = (const float*)d_in[2];
    const float* Wk         = (const float*)d_in[3];
    const float* bk         = (const float*)d_in[4];
    const float* Wv         = (const float*)d_in[5];
    const float* bv         = (const float*)d_in[6];
    const float* bias_table = (const float*)d_in[7];
    float* out = (float*)d_out;

    float* wsq = (float*)d_ws;
    float* wsk = wsq + QKV_ELEMS;
    float* wsv = wsk + QKV_ELEMS;

    // 1024 M-tiles / 8 waves-per-block, 32 N-tiles, 3 matrices.
    qkv_rope_kernel<<<dim3(128, 32, 3), 256, 0, stream>>>(
        hidden, Wq, bq, Wk, bk, Wv, bv, wsq, wsk, wsv);

    // One block per (b, h, q-tile-of-16).
    attn_kernel<<<dim3(B_ * H_ * (S_ / 16)), 256, 0, stream>>>(
        wsq, wsk, wsv, bias_table, out);
}